// GeometryNet_16484084483042
// MI455X (gfx1250) — compile-verified
//
#include <hip/hip_runtime.h>

typedef __attribute__((ext_vector_type(2))) float v2f;
typedef __attribute__((ext_vector_type(4))) float v4f;
typedef __attribute__((ext_vector_type(8))) float v8f;

#define NPTS   32768
#define EMB    64
#define NCLS   50
#define NH     5
#define NWP1   6                         // 5 windows + 1 frame
#define NBOX   (NH * NWP1)               // 30 boxes per town
#define PTS_PER_WAVE  16
#define WAVES_PER_BLK 8
#define PTS_PER_BLK   (PTS_PER_WAVE * WAVES_PER_BLK)  // 128

__global__ __launch_bounds__(256)
void geometry_kernel(const float* __restrict__ data,
                     const float* __restrict__ shape,
                     float* __restrict__ out_contains,
                     float* __restrict__ out_dist)
{
    __shared__ float s_lo[NBOX * EMB];   // 7.5 KB
    __shared__ float s_hi[NBOX * EMB];   // 7.5 KB

    const int c   = blockIdx.y;
    const int tid = threadIdx.x;

    // Precompute lo/hi for all 30 boxes of this town into LDS.
    // shape layout: [c][h][w][corner(2)][d] -> box b = h*6+w, 128 floats per box.
    {
        const float* S = shape + (size_t)c * (NBOX * 2 * EMB);
        for (int i = tid; i < NBOX * EMB; i += 256) {
            int b = i >> 6, d = i & 63;
            float a0 = S[b * 128 + d];
            float a1 = S[b * 128 + 64 + d];
            s_lo[i] = fminf(a0, a1);
            s_hi[i] = fmaxf(a0, a1);
        }
    }
    __syncthreads();

    const int wave = tid >> 5;
    const int lane = tid & 31;
    const int half = lane >> 4;          // 0: K pair {0,1}; 1: K pair {2,3}
    const int o    = half * 2;           // dim offset inside each 4-dim chunk
    const int n0   = blockIdx.x * PTS_PER_BLK + wave * PTS_PER_WAVE;
    const int pt   = n0 + (lane & 15);   // this lane's point (A-matrix row)

    // Preload this lane's point data: dims 4*kc+o, 4*kc+o+1 for kc = 0..15.
    v2f pdat[16];
    {
        const v2f* dp = reinterpret_cast<const v2f*>(data + (size_t)pt * EMB + o);
        #pragma unroll
        for (int kc = 0; kc < 16; ++kc) pdat[kc] = dp[2 * kc];
    }

    const v2f ones = {1.0f, 1.0f};       // B = all-ones: D[m,n] = rowsum(A[m,:]) + C
    float town[8] = {0.f, 0.f, 0.f, 0.f, 0.f, 0.f, 0.f, 0.f};

    for (int h = 0; h < NH; ++h) {
        v8f acc[NWP1];
        {
            v8f z = {};
            #pragma unroll
            for (int w = 0; w < NWP1; ++w) acc[w] = z;
        }

        // sq[w, point] accumulated over 16 K-chunks of 4 dims; 6 independent
        // accumulator chains pipeline the matrix unit.
        #pragma unroll
        for (int kc = 0; kc < 16; ++kc) {
            v2f p = pdat[kc];
            #pragma unroll
            for (int w = 0; w < NWP1; ++w) {
                const int b = h * NWP1 + w;
                const v2f lo = *reinterpret_cast<const v2f*>(&s_lo[b * EMB + kc * 4 + o]);
                const v2f hi = *reinterpret_cast<const v2f*>(&s_hi[b * EMB + kc * 4 + o]);
                float d0 = fmaxf(lo.x - p.x, 0.0f) + fmaxf(p.x - hi.x, 0.0f);
                float d1 = fmaxf(lo.y - p.y, 0.0f) + fmaxf(p.y - hi.y, 0.0f);
                v2f a;
                a.x = d0 * d0;
                a.y = d1 * d1;
                acc[w] = __builtin_amdgcn_wmma_f32_16x16x4_f32(
                    false, a, false, ones, (short)0, acc[w], false, false);
            }
        }

        // dists: lane 0 owns points n0..n0+7 (acc VGPRs 0..7, lanes 0-15),
        //        lane 16 owns points n0+8..n0+15 (lanes 16-31).
        #pragma unroll
        for (int w = 0; w < NWP1; ++w) {
            if (lane == 0 || lane == 16) {
                v8f s = acc[w];
                v4f q0, q1;
                q0.x = sqrtf(s[0]); q0.y = sqrtf(s[1]); q0.z = sqrtf(s[2]); q0.w = sqrtf(s[3]);
                q1.x = sqrtf(s[4]); q1.y = sqrtf(s[5]); q1.z = sqrtf(s[6]); q1.w = sqrtf(s[7]);
                float* dst = out_dist
                           + (size_t)((c * NH + h) * NWP1 + w) * NPTS
                           + n0 + half * 8;
                *reinterpret_cast<v4f*>(dst)     = q0;
                *reinterpret_cast<v4f*>(dst + 4) = q1;
            }
        }

        // inside(box) <=> sq == 0.0 exactly (both relu terms are 0 inside).
        #pragma unroll
        for (int i = 0; i < 8; ++i) {
            bool winany = (acc[0][i] == 0.0f) | (acc[1][i] == 0.0f) |
                          (acc[2][i] == 0.0f) | (acc[3][i] == 0.0f) |
                          (acc[4][i] == 0.0f);
            bool frame  = (acc[5][i] == 0.0f);
            if (frame && !winany) town[i] = 1.0f;
        }
    }

    if (lane == 0 || lane == 16) {
        float* dst = out_contains + (size_t)c * NPTS + n0 + half * 8;
        v4f t0, t1;
        t0.x = town[0]; t0.y = town[1]; t0.z = town[2]; t0.w = town[3];
        t1.x = town[4]; t1.y = town[5]; t1.z = town[6]; t1.w = town[7];
        *reinterpret_cast<v4f*>(dst)     = t0;
        *reinterpret_cast<v4f*>(dst + 4) = t1;
    }
}

extern "C" void kernel_launch(void* const* d_in, const int* in_sizes, int n_in,
                              void* d_out, int out_size, void* d_ws, size_t ws_size,
                              hipStream_t stream) {
    (void)in_sizes; (void)n_in; (void)out_size; (void)d_ws; (void)ws_size;

    const float* data  = (const float*)d_in[0];   // [32768, 64] f32
    const float* shape = (const float*)d_in[1];   // [50, 5, 6, 2, 64] f32
    float* out = (float*)d_out;

    // Output 0: data passthrough (graph-capture-safe async D2D copy).
    hipMemcpyAsync(out, data, (size_t)NPTS * EMB * sizeof(float),
                   hipMemcpyDeviceToDevice, stream);

    float* out_contains = out + (size_t)NPTS * EMB;               // [50, 32768]
    float* out_dist     = out_contains + (size_t)NCLS * NPTS;     // [50, 5, 6, 32768]

    dim3 grid(NPTS / PTS_PER_BLK, NCLS, 1);   // 256 x 50 blocks
    geometry_kernel<<<grid, 256, 0, stream>>>(data, shape, out_contains, out_dist);
}